// AttEncoder_16475494548034
// MI455X (gfx1250) — compile-verified
//
#include <hip/hip_runtime.h>

// Fused AttEncoder for MI455X (gfx1250, wave32).
// conv1d-as-GEMM via chained V_WMMA_F32_16X16X4_F32 (exact fp32), fully fused
// channel attention; feat [B,C,1536,T] (393MB) never touches HBM.

typedef __attribute__((ext_vector_type(2))) float v2f;
typedef __attribute__((ext_vector_type(8))) float v8f;

namespace {
constexpr int Bn     = 4;
constexpr int Cn     = 4;
constexpr int Ln     = 32000;
constexpr int F3     = 1536;
constexpr int Kn     = 16;
constexpr int STRIDE = 8;
constexpr int Nn     = F3 / 3;                    // 512
constexpr int Tn     = (Ln - Kn) / STRIDE + 1;    // 3999
constexpr int TT     = 16;                        // time tile
constexpr int NTILES = (Tn + TT - 1) / TT;        // 250
constexpr int XS_LEN = (TT - 1) * STRIDE + Kn;    // 136 samples per channel
constexpr int XS_PAD = 144;                       // padded LDS row
constexpr int TP     = TT + 1;                    // 17: pad vs bank conflicts
}

__global__ __launch_bounds__(256)
void attenc_fused(const float* __restrict__ x,   // [B, C, L]
                  const float* __restrict__ W,   // [C, F3, 1, K]
                  float* __restrict__ out)       // [B, C, N, T]
{
    __shared__ float xs[Cn][XS_PAD];          // x window per channel
    __shared__ float kq[2][Cn][TT][TP];       // k/q feat tiles [part][ch][n][t]
    __shared__ float vt[Cn][TT][TP];          // v feat tiles
    __shared__ float wlog[Cn][Cn][TT];        // attention logits [i][j][t]
    __shared__ float wsm[Cn][Cn][TT];         // softmaxed weights

    const int blk = blockIdx.x;
    const int b   = blk / NTILES;
    const int t0  = (blk % NTILES) * TT;

    const int tid  = threadIdx.x;
    const int wave = tid >> 5;                // 8 waves
    const int lane = tid & 31;
    const int hl   = lane & 15;               // 0..15 within half-wave
    const int hi   = lane >> 4;               // half-wave select

    // ---- stage x window into LDS (zero-clamped at tail) -------------------
    for (int idx = tid; idx < Cn * XS_PAD; idx += 256) {
        const int c = idx / XS_PAD;
        const int i = idx % XS_PAD;
        const int g = t0 * STRIDE + i;
        float v = 0.0f;
        if (i < XS_LEN && g < Ln) v = x[(b * Cn + c) * Ln + g];
        xs[c][i] = v;
    }
    __syncthreads();

    // ---- per-wave WMMA B fragments (patch matrix, invariant over n) -------
    // wave -> channel wc = wave&3 ; phase-1 part = wave>>2 (0 = k, 1 = q)
    // B chunk kk holds rows K = 4*kk .. 4*kk+3 of the 16 x TT patch matrix:
    //   lanes 0-15 : rows 4kk+0 (.x), 4kk+1 (.y), column = lane
    //   lanes16-31 : rows 4kk+2 (.x), 4kk+3 (.y), column = lane-16
    const int wc = wave & 3;
    v2f bf[4];
#pragma unroll
    for (int kk = 0; kk < 4; ++kk) {
        const int k = 4 * kk + 2 * hi;
        bf[kk].x = xs[wc][hl * STRIDE + k + 0];
        bf[kk].y = xs[wc][hl * STRIDE + k + 1];
    }

    // ---- phase 1: accumulate attention logits over all 32 n-chunks --------
    const int wi = tid >> 6;          // query channel owned for logit accum
    const int wj = (tid >> 4) & 3;    // key channel
    const int wt = tid & 15;          // time slot
    const int part = wave >> 2;       // 0 = k rows, 1 = q rows of W

    float wacc = 0.0f;
    for (int n0 = 0; n0 < Nn; n0 += TT) {
        // A fragments: W rows f = part*512 + n0 + hl, chunked along K.
        const float* wrow =
            W + ((size_t)(wc * F3 + part * Nn + n0 + hl)) * Kn + 2 * hi;
        v8f acc = {0.f, 0.f, 0.f, 0.f, 0.f, 0.f, 0.f, 0.f};
#pragma unroll
        for (int kk = 0; kk < 4; ++kk) {
            v2f af;
            af.x = wrow[4 * kk + 0];
            af.y = wrow[4 * kk + 1];
            acc = __builtin_amdgcn_wmma_f32_16x16x4_f32(
                false, af, false, bf[kk], (short)0, acc, false, false);
        }
        // D layout: vgpr r -> row (r + 8*hi), column hl
#pragma unroll
        for (int r = 0; r < 8; ++r)
            kq[part][wc][8 * hi + r][hl] = acc[r];
        __syncthreads();

        // logits: one (i,j,t) per thread, reduce the 16 n's of this chunk
#pragma unroll
        for (int nn = 0; nn < TT; ++nn)
            wacc += kq[1][wi][nn][wt] * kq[0][wj][nn][wt];
        __syncthreads();   // protect kq before next chunk overwrites it
    }
    wlog[wi][wj][wt] = wacc;
    __syncthreads();

    // ---- softmax over key channels (j), 64 active threads -----------------
    if (tid < Cn * TT) {
        const int i = tid >> 4;
        const int t = tid & 15;
        float m = wlog[i][0][t];
#pragma unroll
        for (int j = 1; j < Cn; ++j) m = fmaxf(m, wlog[i][j][t]);
        float e[Cn], s = 0.0f;
#pragma unroll
        for (int j = 0; j < Cn; ++j) { e[j] = __expf(wlog[i][j][t] - m); s += e[j]; }
        const float inv = 1.0f / s;
#pragma unroll
        for (int j = 0; j < Cn; ++j) wsm[i][j][t] = e[j] * inv;
    }
    __syncthreads();

    // ---- phase 2: v tiles + weighted mix, streamed to HBM -----------------
    const int og = tid >> 4;                   // 0..15
    const int ot = tid & 15;                   // time slot (coalesced stores)
    const bool tvalid = (t0 + ot) < Tn;

    for (int n0 = 0; n0 < Nn; n0 += TT) {
        if (wave < 4) {                        // waves 0-3: one v tile each
            const float* wrow =
                W + ((size_t)(wc * F3 + 2 * Nn + n0 + hl)) * Kn + 2 * hi;
            v8f acc = {0.f, 0.f, 0.f, 0.f, 0.f, 0.f, 0.f, 0.f};
#pragma unroll
            for (int kk = 0; kk < 4; ++kk) {
                v2f af;
                af.x = wrow[4 * kk + 0];
                af.y = wrow[4 * kk + 1];
                acc = __builtin_amdgcn_wmma_f32_16x16x4_f32(
                    false, af, false, bf[kk], (short)0, acc, false, false);
            }
#pragma unroll
            for (int r = 0; r < 8; ++r)
                vt[wc][8 * hi + r][hl] = acc[r];
        }
        __syncthreads();

        // out[b,i,n0+nn,t0+ot] = 0.5*(sum_j wsm[i][j][t]*v[j][nn][t] + v[i][nn][t])
#pragma unroll
        for (int e2 = 0; e2 < 4; ++e2) {
            const int idx = og + 16 * e2;      // 0..63
            const int i   = idx >> 4;
            const int nn  = idx & 15;
            float s = 0.0f;
#pragma unroll
            for (int j = 0; j < Cn; ++j) s += wsm[i][j][ot] * vt[j][nn][ot];
            s = 0.5f * (s + vt[i][nn][ot]);
            if (tvalid)
                out[(((size_t)(b * Cn + i)) * Nn + (n0 + nn)) * Tn + (t0 + ot)] = s;
        }
        __syncthreads();   // protect vt before next chunk
    }
}

extern "C" void kernel_launch(void* const* d_in, const int* in_sizes, int n_in,
                              void* d_out, int out_size, void* d_ws, size_t ws_size,
                              hipStream_t stream) {
    (void)in_sizes; (void)n_in; (void)d_ws; (void)ws_size; (void)out_size;
    const float* x = (const float*)d_in[0];   // [4, 4, 32000] f32
    const float* W = (const float*)d_in[1];   // [4, 1536, 1, 16] f32
    float* out = (float*)d_out;               // [4, 4, 512, 3999] f32

    dim3 grid(Bn * NTILES);                   // 1000 blocks
    dim3 block(256);                          // 8 wave32
    attenc_fused<<<grid, block, 0, stream>>>(x, W, out);
}